// SSL_88845693485291
// MI455X (gfx1250) — compile-verified
//
#include <hip/hip_runtime.h>
#include <hip/hip_bf16.h>

// ---------------------------------------------------------------------------
// CDNA5 (gfx1250) implementation of the SSL reference.
// All matrix math uses V_WMMA_F32_16X16X4_F32 (native f32 WMMA, wave32).
// Sequential scan is parallelized as a 3-phase chunked linear recurrence.
// GEMM supports a transB mode: B stored as B[n][k] so each WMMA B-fragment
// is one contiguous v2f (global_load_b64).  All WMMA k-loops are software-
// pipelined one stage deep so fragment loads for step k+4 are in flight
// while step k's WMMAs execute (kills the s_wait_loadcnt-before-every-wmma
// pattern seen in the previous round's assembly).
// ---------------------------------------------------------------------------

typedef __attribute__((ext_vector_type(2))) float v2f;
typedef __attribute__((ext_vector_type(8))) float v8f;

#define D_STATE 512
#define D_IN    256
#define D_OUT   256
#define BATCH   16
#define SEQ     4096
#define NCHUNK  64
#define CHUNK   64

__device__ __forceinline__ v8f wmma4(v2f a, v2f b, v8f c) {
  // D = A(16x4, f32) x B(4x16, f32) + C(16x16, f32)
  return __builtin_amdgcn_wmma_f32_16x16x4_f32(
      /*neg_a=*/false, a, /*neg_b=*/false, b,
      /*c_mod=*/(short)0, c, /*reuse_a=*/false, /*reuse_b=*/false);
}

// ---------------------------------------------------------------------------
// Power iteration for sigma = ||K_raw||_2  (single workgroup, fixed iters)
// writes scale = 1/(max(sigma,1e-5)+0.002)
// ---------------------------------------------------------------------------
__global__ void spectral_scale_kernel(const float* __restrict__ Kraw,
                                      float* __restrict__ scale_out) {
  const int N = 768;
  __shared__ float v[768];
  __shared__ float wv[768];
  __shared__ float red[256];
  int tid = threadIdx.x;
  for (int i = tid; i < N; i += 256) v[i] = 1.0f;
  __syncthreads();
  for (int iter = 0; iter < 64; ++iter) {
    // wv = K v
    for (int r = tid; r < N; r += 256) {
      float s = 0.f;
      const float* row = Kraw + (size_t)r * N;
      for (int c = 0; c < N; ++c) s += row[c] * v[c];
      wv[r] = s;
    }
    __syncthreads();
    // v = K^T wv   (all reads of old v completed at the barrier above)
    for (int c = tid; c < N; c += 256) {
      float s = 0.f;
      for (int r = 0; r < N; ++r) s += Kraw[(size_t)r * N + c] * wv[r];
      v[c] = s;
    }
    __syncthreads();
    // normalize v
    float ss = 0.f;
    for (int i = tid; i < N; i += 256) ss += v[i] * v[i];
    red[tid] = ss;
    __syncthreads();
    for (int off = 128; off > 0; off >>= 1) {
      if (tid < off) red[tid] += red[tid + off];
      __syncthreads();
    }
    float inv = rsqrtf(red[0] + 1e-30f);
    for (int i = tid; i < N; i += 256) v[i] *= inv;
    __syncthreads();
  }
  // sigma = ||K v||
  for (int r = tid; r < N; r += 256) {
    float s = 0.f;
    const float* row = Kraw + (size_t)r * N;
    for (int c = 0; c < N; ++c) s += row[c] * v[c];
    wv[r] = s;
  }
  __syncthreads();
  float ss = 0.f;
  for (int i = tid; i < N; i += 256) ss += wv[i] * wv[i];
  red[tid] = ss;
  __syncthreads();
  for (int off = 128; off > 0; off >>= 1) {
    if (tid < off) red[tid] += red[tid + off];
    __syncthreads();
  }
  if (tid == 0) {
    float sigma = sqrtf(red[0]);
    sigma = fmaxf(sigma, 1e-5f);
    scale_out[0] = 1.0f / (sigma + 0.002f);
  }
}

// K = K_raw * scale
__global__ void scale_kernel(const float* __restrict__ in, float* __restrict__ out,
                             const float* __restrict__ scale, size_t n) {
  float s = scale[0];
  for (size_t i = blockIdx.x * 256ull + threadIdx.x; i < n; i += gridDim.x * 256ull)
    out[i] = in[i] * s;
}

__global__ void identity_kernel(float* __restrict__ X, int n) {
  size_t total = (size_t)n * n;
  for (size_t i = blockIdx.x * 256ull + threadIdx.x; i < total; i += gridDim.x * 256ull) {
    int r = (int)(i / n), c = (int)(i % n);
    X[i] = (r == c) ? 1.0f : 0.0f;
  }
}

// states[b,0,:] = state_in
__global__ void init_state_row_kernel(const float* __restrict__ st, float* __restrict__ states) {
  int i = blockIdx.x * 256 + threadIdx.x;  // 0 .. 16*512-1
  if (i < BATCH * D_STATE) {
    int b = i / D_STATE, n = i % D_STATE;
    states[((size_t)b * (SEQ + 1)) * D_STATE + n] = st[i];
  }
}

// ---------------------------------------------------------------------------
// Generic f32 WMMA GEMM:
//   transB==0:  Cout = alpha * (A @ B)   + beta * Cin   (B row-major [K][N])
//   transB==1:  Cout = alpha * (A @ B^T) + beta * Cin   (B row-major [N][K])
// Workgroup tile 64(M) x 128(N); 8 waves; wave tile 16x64 (4 WMMA C tiles).
// Requires M%64==0, N%128==0, K%4==0 (true for all launches here).
// blockIdx.z batches with the given strides.
// ---------------------------------------------------------------------------
__global__ void gemm_f32_kernel(const float* __restrict__ A, const float* __restrict__ B,
                                const float* __restrict__ Cin, float* __restrict__ Cout,
                                int M, int N, int K, int lda, int ldb, int ldci, int ldco,
                                float alpha, float beta, int transB,
                                long long sA, long long sB, long long sCi, long long sCo) {
  A += (long long)blockIdx.z * sA;
  B += (long long)blockIdx.z * sB;
  Cin += (long long)blockIdx.z * sCi;
  Cout += (long long)blockIdx.z * sCo;

  int tid = threadIdx.x;
  int w = tid >> 5, lane = tid & 31;
  int l15 = lane & 15, lh = lane >> 4;  // half-wave select
  int m0 = blockIdx.y * 64 + (w & 3) * 16;
  int n0 = blockIdx.x * 128 + (w >> 2) * 64;

  v8f acc[4];
  v8f z = {};
#pragma unroll
  for (int j = 0; j < 4; ++j) acc[j] = z;

  const float* Arow = A + (size_t)(m0 + l15) * lda + lh * 2;
  if (transB) {
    // B fragment: one contiguous v2f per tile per k-step (global_load_b64),
    // software-pipelined one k-step ahead.
    const float* Bn[4];
#pragma unroll
    for (int j = 0; j < 4; ++j)
      Bn[j] = B + (size_t)(n0 + j * 16 + l15) * ldb + lh * 2;
    v2f a_cur = *(const v2f*)(Arow);
    v2f b_cur[4];
#pragma unroll
    for (int j = 0; j < 4; ++j) b_cur[j] = *(const v2f*)(Bn[j]);
    for (int k0 = 0; k0 < K - 4; k0 += 4) {
      v2f a_nxt = *(const v2f*)(Arow + k0 + 4);
      v2f b_nxt[4];
#pragma unroll
      for (int j = 0; j < 4; ++j) b_nxt[j] = *(const v2f*)(Bn[j] + k0 + 4);
#pragma unroll
      for (int j = 0; j < 4; ++j) acc[j] = wmma4(a_cur, b_cur[j], acc[j]);
      a_cur = a_nxt;
#pragma unroll
      for (int j = 0; j < 4; ++j) b_cur[j] = b_nxt[j];
    }
#pragma unroll
    for (int j = 0; j < 4; ++j) acc[j] = wmma4(a_cur, b_cur[j], acc[j]);
  } else {
    v2f a_cur = *(const v2f*)(Arow);
    v2f b_cur[4];
    {
      const float* Bk = B + (size_t)(lh * 2) * ldb;
#pragma unroll
      for (int j = 0; j < 4; ++j) {
        int n = n0 + j * 16 + l15;
        b_cur[j].x = Bk[n];
        b_cur[j].y = Bk[ldb + n];
      }
    }
    for (int k0 = 0; k0 < K - 4; k0 += 4) {
      v2f a_nxt = *(const v2f*)(Arow + k0 + 4);
      const float* Bk = B + (size_t)(k0 + 4 + lh * 2) * ldb;
      v2f b_nxt[4];
#pragma unroll
      for (int j = 0; j < 4; ++j) {
        int n = n0 + j * 16 + l15;
        b_nxt[j].x = Bk[n];
        b_nxt[j].y = Bk[ldb + n];
      }
#pragma unroll
      for (int j = 0; j < 4; ++j) acc[j] = wmma4(a_cur, b_cur[j], acc[j]);
      a_cur = a_nxt;
#pragma unroll
      for (int j = 0; j < 4; ++j) b_cur[j] = b_nxt[j];
    }
#pragma unroll
    for (int j = 0; j < 4; ++j) acc[j] = wmma4(a_cur, b_cur[j], acc[j]);
  }

  if (beta != 0.0f) {
#pragma unroll
    for (int j = 0; j < 4; ++j) {
      int n = n0 + j * 16 + l15;
#pragma unroll
      for (int r = 0; r < 8; ++r) {
        size_t row = m0 + (lh ? r + 8 : r);
        Cout[row * ldco + n] = alpha * acc[j][r] + beta * Cin[row * ldci + n];
      }
    }
  } else {
#pragma unroll
    for (int j = 0; j < 4; ++j) {
      int n = n0 + j * 16 + l15;
#pragma unroll
      for (int r = 0; r < 8; ++r) {
        size_t row = m0 + (lh ? r + 8 : r);
        Cout[row * ldco + n] = alpha * acc[j][r];
      }
    }
  }
}

// ---------------------------------------------------------------------------
// Chunked scan: one workgroup per chunk; running 16x512 state lives in LDS.
// Each step:  x = x @ A^T + Bu[t]     (16x512 = 16x512 @ 512x512 + 16x512)
// The B operand (A^T) is read transB-style straight from row-major Amat:
// fragment (k, n) = Amat[n][k]  ->  contiguous v2f per lane, pipelined.
// mode 0: init x=0, write only final local state to Ylast[c]
// mode 1: init x=E[c], write every step to states output (rows t+1)
// ---------------------------------------------------------------------------
__global__ void scan_chunks_kernel(const float* __restrict__ Bu, const float* __restrict__ Amat,
                                   const float* __restrict__ E, float* __restrict__ Ylast,
                                   float* __restrict__ statesOut, int mode) {
  __shared__ __align__(16) float xs[BATCH * D_STATE];  // 32 KB
  int c = blockIdx.x;
  int tid = threadIdx.x;
  int w = tid >> 5, lane = tid & 31;
  int l15 = lane & 15, lh = lane >> 4;
  int n0 = w * 64;  // 8 waves cover N = 512

  if (mode == 0) {
    for (int i = tid; i < BATCH * D_STATE; i += 256) xs[i] = 0.0f;
  } else {
    const float* e = E + (size_t)c * BATCH * D_STATE;
    for (int i = tid; i < BATCH * D_STATE; i += 256) xs[i] = e[i];
  }
  __syncthreads();

  const float* An[4];
#pragma unroll
  for (int j = 0; j < 4; ++j)
    An[j] = Amat + (size_t)(n0 + j * 16 + l15) * D_STATE + lh * 2;
  const float* xrow = &xs[l15 * D_STATE + lh * 2];

  for (int t = 0; t < CHUNK; ++t) {
    int tg = c * CHUNK + t;
    v8f acc[4];
    // acc = Bu[:, tg, n-tile]
#pragma unroll
    for (int j = 0; j < 4; ++j) {
      int n = n0 + j * 16 + l15;
#pragma unroll
      for (int r = 0; r < 8; ++r) {
        int m = lh ? r + 8 : r;
        acc[j][r] = Bu[((size_t)m * SEQ + tg) * D_STATE + n];
      }
    }
    // acc += x @ A^T   (software-pipelined one k-step ahead)
    v2f a_cur = *(const v2f*)(xrow);
    v2f b_cur[4];
#pragma unroll
    for (int j = 0; j < 4; ++j) b_cur[j] = *(const v2f*)(An[j]);
    for (int k0 = 0; k0 < D_STATE - 4; k0 += 4) {
      v2f a_nxt = *(const v2f*)(xrow + k0 + 4);
      v2f b_nxt[4];
#pragma unroll
      for (int j = 0; j < 4; ++j) b_nxt[j] = *(const v2f*)(An[j] + k0 + 4);
#pragma unroll
      for (int j = 0; j < 4; ++j) acc[j] = wmma4(a_cur, b_cur[j], acc[j]);
      a_cur = a_nxt;
#pragma unroll
      for (int j = 0; j < 4; ++j) b_cur[j] = b_nxt[j];
    }
#pragma unroll
    for (int j = 0; j < 4; ++j) acc[j] = wmma4(a_cur, b_cur[j], acc[j]);

    __syncthreads();  // all reads of old x complete
#pragma unroll
    for (int j = 0; j < 4; ++j) {
      int n = n0 + j * 16 + l15;
#pragma unroll
      for (int r = 0; r < 8; ++r) {
        int m = lh ? r + 8 : r;
        xs[m * D_STATE + n] = acc[j][r];
        if (mode == 1)
          __builtin_nontemporal_store(
              acc[j][r], &statesOut[((size_t)m * (SEQ + 1) + (tg + 1)) * D_STATE + n]);
      }
    }
    __syncthreads();
  }

  if (mode == 0) {
    float* y = Ylast + (size_t)c * BATCH * D_STATE;
    for (int i = tid; i < BATCH * D_STATE; i += 256) y[i] = xs[i];
  }
}

// ---------------------------------------------------------------------------
// Phase 2: serial chunk-state propagation (single workgroup):
//   E[c] = e;  e = e @ (A^T)^64 + Ylast[c]
// Q = A^64 (row-major); the B operand (A^T)^64 = Q^T read transB-style.
// ---------------------------------------------------------------------------
__global__ void scan_combine_kernel(const float* __restrict__ state_in,
                                    const float* __restrict__ Q,
                                    const float* __restrict__ Ylast,
                                    float* __restrict__ E) {
  __shared__ __align__(16) float xs[BATCH * D_STATE];
  int tid = threadIdx.x;
  int w = tid >> 5, lane = tid & 31;
  int l15 = lane & 15, lh = lane >> 4;
  int n0 = w * 64;

  for (int i = tid; i < BATCH * D_STATE; i += 256) xs[i] = state_in[i];
  __syncthreads();

  const float* Qn[4];
#pragma unroll
  for (int j = 0; j < 4; ++j)
    Qn[j] = Q + (size_t)(n0 + j * 16 + l15) * D_STATE + lh * 2;
  const float* xrow = &xs[l15 * D_STATE + lh * 2];

  for (int c = 0; c < NCHUNK; ++c) {
    float* ec = E + (size_t)c * BATCH * D_STATE;
    for (int i = tid; i < BATCH * D_STATE; i += 256) ec[i] = xs[i];
    __syncthreads();

    v8f acc[4];
    const float* y = Ylast + (size_t)c * BATCH * D_STATE;
#pragma unroll
    for (int j = 0; j < 4; ++j) {
      int n = n0 + j * 16 + l15;
#pragma unroll
      for (int r = 0; r < 8; ++r) {
        int m = lh ? r + 8 : r;
        acc[j][r] = y[m * D_STATE + n];
      }
    }
    v2f a_cur = *(const v2f*)(xrow);
    v2f b_cur[4];
#pragma unroll
    for (int j = 0; j < 4; ++j) b_cur[j] = *(const v2f*)(Qn[j]);
    for (int k0 = 0; k0 < D_STATE - 4; k0 += 4) {
      v2f a_nxt = *(const v2f*)(xrow + k0 + 4);
      v2f b_nxt[4];
#pragma unroll
      for (int j = 0; j < 4; ++j) b_nxt[j] = *(const v2f*)(Qn[j] + k0 + 4);
#pragma unroll
      for (int j = 0; j < 4; ++j) acc[j] = wmma4(a_cur, b_cur[j], acc[j]);
      a_cur = a_nxt;
#pragma unroll
      for (int j = 0; j < 4; ++j) b_cur[j] = b_nxt[j];
    }
#pragma unroll
    for (int j = 0; j < 4; ++j) acc[j] = wmma4(a_cur, b_cur[j], acc[j]);

    __syncthreads();
#pragma unroll
    for (int j = 0; j < 4; ++j) {
      int n = n0 + j * 16 + l15;
#pragma unroll
      for (int r = 0; r < 8; ++r) {
        int m = lh ? r + 8 : r;
        xs[m * D_STATE + n] = acc[j][r];
      }
    }
    __syncthreads();
  }
}

// ---------------------------------------------------------------------------
extern "C" void kernel_launch(void* const* d_in, const int* in_sizes, int n_in,
                              void* d_out, int out_size, void* d_ws, size_t ws_size,
                              hipStream_t stream) {
  const float* u = (const float*)d_in[0];        // (16,4096,256)
  const float* state = (const float*)d_in[1];    // (16,512)
  const float* S = (const float*)d_in[2];        // (512,512)
  const float* Kraw = (const float*)d_in[3];     // (768,768)

  float* outPtr = (float*)d_out;                               // (16,4096,256)
  float* statesPtr = outPtr + (size_t)BATCH * SEQ * D_OUT;     // (16,4097,512)

  // ---- workspace layout (floats) ----
  constexpr size_t SZ_K = 768ull * 768;
  constexpr size_t SZ_S = 512ull * 512;
  constexpr size_t SZ_B = 512ull * 256;
  constexpr size_t SZ_ST = (size_t)BATCH * D_STATE;
  float* wsf = (float*)d_ws;
  float* scale = wsf;                 // 16
  float* Ks = wsf + 16;               // scaled K (768x768)
  float* XA = Ks + SZ_K;
  float* XB = XA + SZ_S;
  float* T = XB + SZ_S;
  float* tmp1 = T + SZ_S;
  float* Amat = tmp1 + SZ_S;          // A (512,512) row-major
  float* Bmat = Amat + SZ_S;          // B (512,256) row-major  (n, h)
  float* Cmat = Bmat + SZ_B;          // C (256,512) row-major  (o, n)
  float* P0 = Cmat + SZ_B;
  float* P1 = P0 + SZ_S;
  float* Ylast = P1 + SZ_S;           // 64 x (16x512)
  float* E = Ylast + NCHUNK * SZ_ST;  // 64 x (16x512)
  float* Bu = E + NCHUNK * SZ_ST;     // (16*4096, 512)

  auto gemm = [&](const float* A, const float* B, const float* Cin, float* Cout,
                  int M, int N, int K, int lda, int ldb, int ldci, int ldco,
                  float alpha, float beta, int transB, int batch,
                  long long sA, long long sB, long long sCi, long long sCo) {
    dim3 g(N / 128, M / 64, batch);
    gemm_f32_kernel<<<g, 256, 0, stream>>>(A, B, Cin, Cout, M, N, K, lda, ldb, ldci,
                                           ldco, alpha, beta, transB, sA, sB, sCi, sCo);
  };

  // 1) sigma + K scaling
  spectral_scale_kernel<<<1, 256, 0, stream>>>(Kraw, scale);
  scale_kernel<<<1024, 256, 0, stream>>>(Kraw, Ks, scale, SZ_K);

  // 2) Sinv via Newton-Schulz (X0 = I; ||I - S|| ~ 0.45 -> quadratic conv.)
  identity_kernel<<<512, 256, 0, stream>>>(XA, 512);
  float* Xc = XA;
  float* Xn = XB;
  for (int it = 0; it < 8; ++it) {
    gemm(S, Xc, nullptr, T, 512, 512, 512, 512, 512, 0, 512, 1.f, 0.f, 0, 1, 0, 0, 0, 0);
    gemm(Xc, T, Xc, Xn, 512, 512, 512, 512, 512, 512, 512, -1.f, 2.f, 0, 1, 0, 0, 0, 0);
    float* t2 = Xc; Xc = Xn; Xn = t2;
  }
  const float* Sinv = Xc;

  // 3) A = Sinv @ K11 @ S ; B = Sinv @ K12 ; C = K21 @ S ; D = K22 (GAMMA=1)
  const float* K11 = Ks;                       // lda 768
  const float* K12 = Ks + 512;                 // lda 768
  const float* K21 = Ks + 512ull * 768;        // lda 768
  const float* K22 = Ks + 512ull * 768 + 512;  // lda 768
  gemm(K11, S, nullptr, tmp1, 512, 512, 512, 768, 512, 0, 512, 1.f, 0.f, 0, 1, 0, 0, 0, 0);
  gemm(Sinv, tmp1, nullptr, Amat, 512, 512, 512, 512, 512, 0, 512, 1.f, 0.f, 0, 1, 0, 0, 0, 0);
  gemm(Sinv, K12, nullptr, Bmat, 512, 256, 512, 512, 768, 0, 256, 1.f, 0.f, 0, 1, 0, 0, 0, 0);
  gemm(K21, S, nullptr, Cmat, 256, 512, 512, 768, 512, 0, 512, 1.f, 0.f, 0, 1, 0, 0, 0, 0);

  // 4) Bu = u @ B^T      (65536 x 512), B given row-major (n,h) -> transB
  gemm(u, Bmat, nullptr, Bu, BATCH * SEQ, D_STATE, D_IN, D_IN, D_IN, 0, D_STATE,
       1.f, 0.f, 1, 1, 0, 0, 0, 0);

  // 5) Q = A^64 via repeated squaring (row-major powers of A)
  gemm(Amat, Amat, nullptr, P0, 512, 512, 512, 512, 512, 0, 512, 1.f, 0.f, 0, 1, 0, 0, 0, 0);  // ^2
  gemm(P0, P0, nullptr, P1, 512, 512, 512, 512, 512, 0, 512, 1.f, 0.f, 0, 1, 0, 0, 0, 0);      // ^4
  gemm(P1, P1, nullptr, P0, 512, 512, 512, 512, 512, 0, 512, 1.f, 0.f, 0, 1, 0, 0, 0, 0);      // ^8
  gemm(P0, P0, nullptr, P1, 512, 512, 512, 512, 512, 0, 512, 1.f, 0.f, 0, 1, 0, 0, 0, 0);      // ^16
  gemm(P1, P1, nullptr, P0, 512, 512, 512, 512, 512, 0, 512, 1.f, 0.f, 0, 1, 0, 0, 0, 0);      // ^32
  gemm(P0, P0, nullptr, P1, 512, 512, 512, 512, 512, 0, 512, 1.f, 0.f, 0, 1, 0, 0, 0, 0);      // ^64

  // 6) 3-phase chunked scan (A^T operand read transB-style from Amat / Q)
  scan_chunks_kernel<<<NCHUNK, 256, 0, stream>>>(Bu, Amat, nullptr, Ylast, nullptr, 0);
  scan_combine_kernel<<<1, 256, 0, stream>>>(state, P1, Ylast, E);
  scan_chunks_kernel<<<NCHUNK, 256, 0, stream>>>(Bu, Amat, E, nullptr, statesPtr, 1);
  init_state_row_kernel<<<(BATCH * D_STATE + 255) / 256, 256, 0, stream>>>(state, statesPtr);

  // 7) out = pre_states @ C^T + u @ D^T   (C, D row-major -> transB)
  gemm(statesPtr, Cmat, nullptr, outPtr, SEQ, D_OUT, D_STATE, D_STATE, D_STATE, 0, D_OUT,
       1.f, 0.f, 1, BATCH,
       (long long)(SEQ + 1) * D_STATE, 0, 0, (long long)SEQ * D_OUT);
  gemm(u, K22, outPtr, outPtr, BATCH * SEQ, D_OUT, D_IN, D_IN, 768, D_OUT, D_OUT,
       1.f, 1.f, 1, 1, 0, 0, 0, 0);
}